// Model_17910013624325
// MI455X (gfx1250) — compile-verified
//
#include <hip/hip_runtime.h>

typedef __attribute__((ext_vector_type(2))) float v2f;
typedef __attribute__((ext_vector_type(4))) float v4f;
typedef __attribute__((ext_vector_type(8))) float v8f;

#define NPAD 30      // input feature length
#define H1   60      // hidden width
#define NOUT 5       // embedding width
#define ESTR 8       // padded embedding stride (slots 5..7 zeroed)
#define NROWS 4096

// ---------------------------------------------------------------------------
// Embedding kernel: emb = relu(relu(X @ W1 + b1) @ W2 + b2), emitted padded
// to stride 8 with zeros. One wave computes 16 rows via V_WMMA_F32_16X16X4_F32.
// Block = 256 threads = 8 waves = 128 rows; grid = 32 blocks = 4096 rows.
// ---------------------------------------------------------------------------
__global__ __launch_bounds__(256) void embed_kernel(
    const float* __restrict__ X,    // [4096, 30]
    const float* __restrict__ W1,   // [30, 60]
    const float* __restrict__ b1,   // [60]
    const float* __restrict__ W2,   // [60, 5]
    const float* __restrict__ b2,   // [5]
    float* __restrict__ emb)        // [4096, 8]
{
    __shared__ float W1p[32][64];      // zero-padded W1
    __shared__ float W2p[64][16];      // zero-padded W2
    __shared__ float b1p[64];
    __shared__ float b2p[16];
    __shared__ float Hp[8][16][64];    // per-wave hidden activations (zero-padded)

    const int tid = threadIdx.x;

    // Cooperative zero-padded weight staging.
#pragma unroll
    for (int i = 0; i < 8; ++i) {
        int idx = tid + i * 256;               // 2048 entries
        int k = idx >> 6, n = idx & 63;
        W1p[k][n] = (k < NPAD && n < H1) ? W1[k * H1 + n] : 0.0f;
    }
#pragma unroll
    for (int i = 0; i < 4; ++i) {
        int idx = tid + i * 256;               // 1024 entries
        int k = idx >> 4, n = idx & 15;
        W2p[k][n] = (k < H1 && n < NOUT) ? W2[k * NOUT + n] : 0.0f;
    }
    if (tid < 64) b1p[tid] = (tid < H1) ? b1[tid] : 0.0f;
    if (tid < 16) b2p[tid] = (tid < NOUT) ? b2[tid] : 0.0f;
    __syncthreads();

    const int wave = tid >> 5;
    const int lane = tid & 31;
    const int r    = lane & 15;   // row (A) / column (B,C,D) index for this lane
    const int hi   = lane >> 4;   // 0: K pair {0,1}; 1: K pair {2,3}
    const int rowbase = blockIdx.x * 128 + wave * 16;

    // ---- Layer 1: H[16x60] = relu(X[16x30] @ W1 + b1), N in 4 tiles of 16 ----
    const float* xrow = X + (size_t)(rowbase + r) * NPAD;  // lane's input row
#pragma unroll
    for (int nt = 0; nt < 4; ++nt) {
        v8f acc = {};
#pragma unroll
        for (int s = 0; s < 8; ++s) {          // K = 32 (30 real + 2 zero)
            int k0 = 4 * s + 2 * hi;
            v2f a, b;
            a.x = (k0     < NPAD) ? xrow[k0]     : 0.0f;
            a.y = (k0 + 1 < NPAD) ? xrow[k0 + 1] : 0.0f;
            b.x = W1p[k0][nt * 16 + r];        // rows 30,31 are zero padding
            b.y = W1p[k0 + 1][nt * 16 + r];
            acc = __builtin_amdgcn_wmma_f32_16x16x4_f32(
                false, a, false, b, (short)0, acc, false, false);
        }
        // acc: this lane holds column (nt*16 + r) of rows v + 8*hi
        const int col = nt * 16 + r;
        const float bias = b1p[col & 63];
#pragma unroll
        for (int v = 0; v < 8; ++v) {
            float h = acc[v] + bias;
            h = (col < H1) ? (h > 0.0f ? h : 0.0f) : 0.0f;
            Hp[wave][v + 8 * hi][col & 63] = h;
        }
    }
    __syncthreads();   // make each wave's Hp writes visible to all its lanes

    // ---- Layer 2: E[16x5] = relu(H[16x64] @ W2p[64x16] + b2), one N tile ----
    v8f acc2 = {};
#pragma unroll
    for (int s = 0; s < 16; ++s) {             // K = 64 (60 real + 4 zero)
        int k0 = 4 * s + 2 * hi;
        v2f a, b;
        a.x = Hp[wave][r][k0];
        a.y = Hp[wave][r][k0 + 1];
        b.x = W2p[k0][r];
        b.y = W2p[k0 + 1][r];
        acc2 = __builtin_amdgcn_wmma_f32_16x16x4_f32(
            false, a, false, b, (short)0, acc2, false, false);
    }
    // Lane holds column r of rows v + 8*hi; emit stride-8 padded embedding.
    if (r < ESTR) {
        const float bias = b2p[r];
#pragma unroll
        for (int v = 0; v < 8; ++v) {
            float e = acc2[v] + bias;
            e = (e > 0.0f) ? e : 0.0f;
            emb[(size_t)(rowbase + v + 8 * hi) * ESTR + r] = (r < NOUT) ? e : 0.0f;
        }
    }
}

// ---------------------------------------------------------------------------
// Scoring kernel: out[i,j] = -sum_k relu(qe[i,k] - ce[j,k]).
// Store-bandwidth-bound (67 MB output). 64q x 128c tile per 256-thread block;
// each thread computes an 8q x 4c micro-tile and writes float4 (b128) stores.
// ---------------------------------------------------------------------------
__global__ __launch_bounds__(256) void score_kernel(
    const float* __restrict__ qe,   // [4096, 8]
    const float* __restrict__ ce,   // [4096, 8]
    float* __restrict__ out)        // [4096, 4096]
{
    __shared__ float qs[64][ESTR];
    __shared__ float cs[128][ESTR];

    const int tid   = threadIdx.x;
    const int cbase = blockIdx.x * 128;
    const int qbase = blockIdx.y * 64;

    // Vectorized tile staging.
    const v4f* qg = (const v4f*)(qe + (size_t)qbase * ESTR);   // 128 x v4f
    const v4f* cg = (const v4f*)(ce + (size_t)cbase * ESTR);   // 256 x v4f
    v4f* qsv = (v4f*)&qs[0][0];
    v4f* csv = (v4f*)&cs[0][0];
    if (tid < 128) qsv[tid] = qg[tid];
    csv[tid] = cg[tid];
    __syncthreads();

    const int tx = tid & 31;   // corpus group: 4 consecutive j
    const int ty = tid >> 5;   // query group: 8 consecutive i

    // Cache this thread's 4 corpus embeddings in registers.
    float c[4][NOUT];
#pragma unroll
    for (int j = 0; j < 4; ++j)
#pragma unroll
        for (int k = 0; k < NOUT; ++k)
            c[j][k] = cs[tx * 4 + j][k];

#pragma unroll
    for (int qi = 0; qi < 8; ++qi) {
        const int row = qbase + ty * 8 + qi;
        float q[NOUT];
#pragma unroll
        for (int k = 0; k < NOUT; ++k) q[k] = qs[ty * 8 + qi][k];

        v4f res;
#pragma unroll
        for (int j = 0; j < 4; ++j) {
            float s = 0.0f;
#pragma unroll
            for (int k = 0; k < NOUT; ++k) {
                float d = q[k] - c[j][k];
                s += (d > 0.0f) ? d : 0.0f;
            }
            res[j] = -s;
        }
        *(v4f*)(out + (size_t)row * NROWS + cbase + tx * 4) = res;
    }
}

extern "C" void kernel_launch(void* const* d_in, const int* in_sizes, int n_in,
                              void* d_out, int out_size, void* d_ws, size_t ws_size,
                              hipStream_t stream) {
    const float* queries = (const float*)d_in[0];  // [4096,30]
    const float* corpus  = (const float*)d_in[1];  // [4096,30]
    const float* W1      = (const float*)d_in[2];  // [30,60]
    const float* b1      = (const float*)d_in[3];  // [60]
    const float* W2      = (const float*)d_in[4];  // [60,5]
    const float* b2      = (const float*)d_in[5];  // [5]
    float* out = (float*)d_out;                    // [4096,4096]

    float* qe = (float*)d_ws;                      // [4096,8] padded embeddings
    float* ce = qe + (size_t)NROWS * ESTR;         // [4096,8]

    embed_kernel<<<32, 256, 0, stream>>>(queries, W1, b1, W2, b2, qe);
    embed_kernel<<<32, 256, 0, stream>>>(corpus,  W1, b1, W2, b2, ce);

    dim3 grid(NROWS / 128, NROWS / 64);            // 32 x 64 blocks
    score_kernel<<<grid, 256, 0, stream>>>(qe, ce, out);
}